// AssemblyGNN_69286412419336
// MI455X (gfx1250) — compile-verified
//
#include <hip/hip_runtime.h>
#include <hip/hip_bf16.h>

typedef float v2f __attribute__((ext_vector_type(2)));
typedef float v8f __attribute__((ext_vector_type(8)));

#define DFEAT 64

// ---------------------------------------------------------------------------
// Degree / normalization
// ---------------------------------------------------------------------------
__global__ void deg_init_kernel(float* __restrict__ deg, int n) {
    int t = blockIdx.x * blockDim.x + threadIdx.x;
    if (t < n) deg[t] = 1.0f;  // self-loop contributes 1 to every node
}

__global__ void deg_accum_kernel(const int* __restrict__ dst,
                                 float* __restrict__ deg, int e) {
    int t = blockIdx.x * blockDim.x + threadIdx.x;
    if (t < e) atomicAdd(&deg[dst[t]], 1.0f);
}

__global__ void dinv_kernel(float* __restrict__ deg, int n) {
    int t = blockIdx.x * blockDim.x + threadIdx.x;
    if (t < n) deg[t] = rsqrtf(deg[t]);  // deg >= 1 always (self-loops)
}

// ---------------------------------------------------------------------------
// Dense GEMM  out[N,64] = act(A)[N,64] @ W[64,64]  using V_WMMA_F32_16X16X4_F32
// One wave computes a 16x64 row panel: 4 column tiles, K=64 in 16 steps of 4.
// RELU_BIAS fuses  act(a) = relu(a + bias[k])  into the A-fragment load
// (layer-2 input is relu(agg1 + b1)).
//
// A 16x4 f32 fragment (ISA 7.12.2): lane L holds row M=L&15; half-wave
// (L>>4) selects K pair {0,1} vs {2,3}; VGPR0=K even, VGPR1=K odd.
// B 4x16 fragment mirrors it with N=L&15.  C/D: VGPR v, lane L ->
// row (L>>4)*8 + v, col L&15.
// ---------------------------------------------------------------------------
template <bool RELU_BIAS>
__global__ void gcn_gemm_kernel(const float* __restrict__ A,
                                const float* __restrict__ W,
                                const float* __restrict__ bias,
                                float* __restrict__ out,
                                int nTiles) {
    const int lane = threadIdx.x & 31;
    const int wave = threadIdx.x >> 5;
    const int tile = blockIdx.x * (blockDim.x >> 5) + wave;
    if (tile >= nTiles) return;  // wave-uniform: EXEC stays all-ones for WMMA

    const int m    = lane & 15;   // A row within tile == B/D column
    const int half = lane >> 4;   // 0 or 1
    const int koff = half << 1;   // K offset: 0 or 2

    const float* __restrict__ arow = A + (size_t)(tile * 16 + m) * DFEAT;

    v8f acc[4] = {v8f{0}, v8f{0}, v8f{0}, v8f{0}};

#pragma unroll
    for (int k = 0; k < DFEAT; k += 4) {
        float a0 = arow[k + koff];
        float a1 = arow[k + koff + 1];
        if (RELU_BIAS) {
            a0 = fmaxf(a0 + bias[k + koff], 0.0f);
            a1 = fmaxf(a1 + bias[k + koff + 1], 0.0f);
        }
        v2f a;
        a.x = a0;
        a.y = a1;
#pragma unroll
        for (int t = 0; t < 4; ++t) {
            v2f b;
            b.x = W[(k + koff) * DFEAT + t * 16 + m];
            b.y = W[(k + koff + 1) * DFEAT + t * 16 + m];
            acc[t] = __builtin_amdgcn_wmma_f32_16x16x4_f32(
                /*neg_a=*/false, a, /*neg_b=*/false, b,
                /*c_mod=*/(short)0, acc[t],
                /*reuse_a=*/false, /*reuse_b=*/false);
        }
    }

    float* __restrict__ orow = out + (size_t)tile * 16 * DFEAT;
#pragma unroll
    for (int t = 0; t < 4; ++t) {
#pragma unroll
        for (int v = 0; v < 8; ++v) {
            orow[(half * 8 + v) * DFEAT + t * 16 + m] = acc[t][v];
        }
    }
}

// ---------------------------------------------------------------------------
// Self-loop contribution: agg[i,:] = dinv[i]^2 * h[i,:]
// ---------------------------------------------------------------------------
__global__ void selfloop_kernel(const float* __restrict__ h,
                                const float* __restrict__ dinv,
                                float* __restrict__ agg, int total) {
    int t = blockIdx.x * blockDim.x + threadIdx.x;
    if (t >= total) return;
    float di = dinv[t >> 6];
    agg[t] = di * di * h[t];
}

// ---------------------------------------------------------------------------
// Edge scatter: agg[dst,:] += dinv[src]*dinv[dst] * h[src,:]
// 4 threads per edge, 16 features each (float4 gathers, f32 atomics).
// ---------------------------------------------------------------------------
__global__ void scatter_kernel(const float* __restrict__ h,
                               const int* __restrict__ src,
                               const int* __restrict__ dst,
                               const float* __restrict__ dinv,
                               float* __restrict__ agg, int nEdges) {
    int t = blockIdx.x * blockDim.x + threadIdx.x;
    int e = t >> 2;
    if (e >= nEdges) return;
    int c = (t & 3) << 4;
    int s = src[e];
    int d = dst[e];
    float nrm = dinv[s] * dinv[d];
    const float4* __restrict__ hp = (const float4*)(h + (size_t)s * DFEAT + c);
    float* __restrict__ ap = agg + (size_t)d * DFEAT + c;
#pragma unroll
    for (int i = 0; i < 4; ++i) {
        float4 v = hp[i];
        atomicAdd(ap + i * 4 + 0, nrm * v.x);
        atomicAdd(ap + i * 4 + 1, nrm * v.y);
        atomicAdd(ap + i * 4 + 2, nrm * v.z);
        atomicAdd(ap + i * 4 + 3, nrm * v.w);
    }
}

// ---------------------------------------------------------------------------
// Output: out[i] = relu(agg2[i,:] + b2) . Wout + bout   (one wave per node)
// ---------------------------------------------------------------------------
__global__ void out_kernel(const float* __restrict__ agg,
                           const float* __restrict__ b2,
                           const float* __restrict__ Wout,
                           const float* __restrict__ bout,
                           float* __restrict__ outp, int n) {
    int lane = threadIdx.x & 31;
    int wave = threadIdx.x >> 5;
    int node = blockIdx.x * (blockDim.x >> 5) + wave;
    if (node >= n) return;
    const float* __restrict__ row = agg + (size_t)node * DFEAT;
    float p = fmaxf(row[lane] + b2[lane], 0.0f) * Wout[lane] +
              fmaxf(row[lane + 32] + b2[lane + 32], 0.0f) * Wout[lane + 32];
#pragma unroll
    for (int off = 16; off > 0; off >>= 1) p += __shfl_down(p, off, 32);
    if (lane == 0) outp[node] = p + bout[0];
}

// ---------------------------------------------------------------------------
// Launch
// ---------------------------------------------------------------------------
extern "C" void kernel_launch(void* const* d_in, const int* in_sizes, int n_in,
                              void* d_out, int out_size, void* d_ws, size_t ws_size,
                              hipStream_t stream) {
    const float* x    = (const float*)d_in[0];
    const int*   ei   = (const int*)d_in[1];   // [2,E] int32: row0=src, row1=dst
    // d_in[2] = batch (unused; whole graph is one batch)
    const float* W1   = (const float*)d_in[3];
    const float* b1   = (const float*)d_in[4];
    const float* W2   = (const float*)d_in[5];
    const float* b2   = (const float*)d_in[6];
    const float* Wout = (const float*)d_in[7];
    const float* bout = (const float*)d_in[8];

    const int N = in_sizes[0] / DFEAT;  // 100000
    const int E = in_sizes[1] / 2;      // 1000000
    const int* src = ei;
    const int* dst = ei + E;

    // Workspace: dinv[N] | bufA[N*64] | bufB[N*64]   (~52 MB)
    float* dinv = (float*)d_ws;
    float* bufA = dinv + N;
    float* bufB = bufA + (size_t)N * DFEAT;

    const int TPB = 256;
    const int nTiles = N / 16;                       // 6250 (N % 16 == 0)
    const int gemmBlocks = (nTiles + 7) / 8;         // 8 waves/block
    const int totalND = N * DFEAT;

    // 1) dinv = rsqrt(deg with self-loops)
    deg_init_kernel<<<(N + TPB - 1) / TPB, TPB, 0, stream>>>(dinv, N);
    deg_accum_kernel<<<(E + TPB - 1) / TPB, TPB, 0, stream>>>(dst, dinv, E);
    dinv_kernel<<<(N + TPB - 1) / TPB, TPB, 0, stream>>>(dinv, N);

    // 2) Layer 1: bufA = x @ W1 ; bufB = aggregate(bufA)
    gcn_gemm_kernel<false><<<gemmBlocks, TPB, 0, stream>>>(x, W1, nullptr, bufA, nTiles);
    selfloop_kernel<<<(totalND + TPB - 1) / TPB, TPB, 0, stream>>>(bufA, dinv, bufB, totalND);
    scatter_kernel<<<(4 * E + TPB - 1) / TPB, TPB, 0, stream>>>(bufA, src, dst, dinv, bufB, E);

    // 3) Layer 2: bufA = relu(bufB + b1) @ W2 ; bufB = aggregate(bufA)
    gcn_gemm_kernel<true><<<gemmBlocks, TPB, 0, stream>>>(bufB, W2, b1, bufA, nTiles);
    selfloop_kernel<<<(totalND + TPB - 1) / TPB, TPB, 0, stream>>>(bufA, dinv, bufB, totalND);
    scatter_kernel<<<(4 * E + TPB - 1) / TPB, TPB, 0, stream>>>(bufA, src, dst, dinv, bufB, E);

    // 4) out = relu(bufB + b2) @ Wout + bout
    out_kernel<<<(N + 7) / 8, TPB, 0, stream>>>(bufB, b2, Wout, bout, (float*)d_out, N);
}